// GraphSSD_80934363725839
// MI455X (gfx1250) — compile-verified
//
#include <hip/hip_runtime.h>
#include <hip/hip_bf16.h>

// ---------------- problem constants (match reference) ----------------
#define NUM_HEADS 4
#define D_INNER   256
#define D_STATE   256
#define HEADDIM   64
#define NUM_GRAPHS 8
#define L_NODES   512
#define N_NODES   (NUM_GRAPHS * L_NODES)   // 4096
#define N_EDGES   65536
#define TOLV      0.05f
#define NMAT      (NUM_GRAPHS * NUM_HEADS) // 32 (b,h) matrices of 512x512
#define MATF      ((size_t)512 * 512)      // floats per matrix
#define BIGF      ((size_t)NMAT * MATF)    // floats per big scratch buffer (8M)

typedef __attribute__((ext_vector_type(2))) float v2f;
typedef __attribute__((ext_vector_type(8))) float v8f;

__device__ __forceinline__ float softplusf(float v) {
  // jax.nn.softplus = log1p(exp(v)), stable for large v
  return (v > 20.0f) ? v : log1pf(expf(v));
}

// V_WMMA_F32_16X16X4_F32 lane fragments (wave32):
//   A (16x4): lane m (0-15): vgpr v -> (M=m, K=v);   lane m+16: (M=m, K=v+2)
//   B (4x16): lane n (0-15): vgpr v -> (K=v, N=n);   lane n+16: (K=v+2, N=n)
//   C/D (16x16): vgpr v: lanes 0-15 -> M=v, lanes 16-31 -> M=v+8; N=lane&15

// ---------------- kernel 1: per-node terms ----------------
__global__ void gssd_prep(const float* __restrict__ dt,
                          const float* __restrict__ dt_bias,
                          float* __restrict__ a0, float* __restrict__ a1,
                          float* __restrict__ dum, float* __restrict__ dself) {
  int i = blockIdx.x * blockDim.x + threadIdx.x;  // i = node*4 + h
  if (i >= N_NODES * NUM_HEADS) return;
  int n = i >> 2, h = i & 3;
  float bh = dt_bias[h];
  const float* p = dt + (size_t)n * 16 + h * 4;
  a0[i]    = -softplusf(p[0] + bh);
  a1[i]    = -softplusf(p[1] + bh);
  dum[i]   = expf(-softplusf(p[2] + bh));
  dself[i] = p[3];
}

// ---------------- kernel 2: edge scatter into adj ----------------
__global__ void gssd_edges(const float* __restrict__ dt_edge,
                           const float* __restrict__ dt_bias,
                           const int* __restrict__ ei,
                           const float* __restrict__ a0,
                           const float* __restrict__ a1,
                           float* __restrict__ adj) {
  int e = blockIdx.x * blockDim.x + threadIdx.x;
  if (e >= N_EDGES) return;
  int src = ei[e];
  int dst = ei[N_EDGES + e];
  int b  = src >> 9;
  int ls = src & 511;
  int ld = dst & 511;
#pragma unroll
  for (int h = 0; h < NUM_HEADS; ++h) {
    float ae = -softplusf(dt_edge[(size_t)e * 4 + h] + dt_bias[h]);
    float v  = expf((a0[src * 4 + h] + a1[dst * 4 + h] + ae) * (1.0f / 3.0f));
    size_t idx = (((size_t)(b * 4 + h)) << 18) + ((size_t)ls << 9) + ld;
    atomicAdd(&adj[idx], v);
  }
}

// ---------------- kernel 3: normalize + transpose -> AT, Q0 = I + AT ----------------
__global__ __launch_bounds__(128)
void gssd_normT(const float* __restrict__ adj, const float* __restrict__ dum,
                float* __restrict__ AT, float* __restrict__ Q0) {
  int i = blockIdx.x;   // row of A (src-local)
  int m = blockIdx.y;   // (b*4+h)
  int t = threadIdx.x;  // 128 threads
  const float* row = adj + (((size_t)m) << 18) + ((size_t)i << 9);
  __shared__ float red[128];
  __shared__ float scale_s;
  red[t] = row[t] + row[t + 128] + row[t + 256] + row[t + 384];
  __syncthreads();
  for (int off = 64; off > 0; off >>= 1) {
    if (t < off) red[t] += red[t + off];
    __syncthreads();
  }
  if (t == 0) {
    int b = m >> 2, h = m & 3;
    float d = dum[(((b << 9) + i) << 2) + h];
    float norm = fmaxf(1.0f, red[0] + d);
    scale_s = 1.0f / (norm + TOLV);
  }
  __syncthreads();
  float sc = scale_s;
#pragma unroll
  for (int q = 0; q < 4; ++q) {
    int j = t + q * 128;
    float v = row[j] * sc;
    size_t o = (((size_t)m) << 18) + ((size_t)j << 9) + i;  // transposed
    AT[o] = v;
    Q0[o] = v + ((i == j) ? 1.0f : 0.0f);
  }
}

// ---------------- kernel 4: batched 512^3 fp32 WMMA GEMM, pipelined ----------------
// D = X @ Y (+ X if addX) over NMAT batched matrices (row-major, ld = 512)
// 128 threads / 4 waves; block tile 64x64; Kc = 32; double-buffered LDS.
// Padded LDS strides: A-tile stride 36 (lp*36 mod 64 bijective -> conflict-free
// b64 fragment loads), B-tile stride 80 (lane halves offset by 32 banks).
#define XSTR 36
#define YSTR 80
__global__ __launch_bounds__(128)
void gssd_gemm512(const float* __restrict__ X, const float* __restrict__ Y,
                  float* __restrict__ Dst, int addX) {
  __shared__ float Xs[2][64 * XSTR];
  __shared__ float Ys[2][32 * YSTR];
  const int t    = threadIdx.x;
  const int wave = t >> 5;
  const int lane = t & 31;
  const int lp   = lane & 15;
  const int half = lane >> 4;
  const int bm   = blockIdx.x * 64;
  const int bn   = blockIdx.y * 64;
  const size_t mb = ((size_t)blockIdx.z) << 18;
  const float* Xb = X + mb;
  const float* Yb = Y + mb;
  float* Db = Dst + mb;

  // staging coordinates: 4 float4 per thread for each tile
  int xr[4], xc[4], yr[4], yc[4];
#pragma unroll
  for (int q = 0; q < 4; ++q) {
    int flat = (t * 4 + q) * 4;
    xr[q] = flat >> 5;  xc[q] = flat & 31;   // X tile: 64 x 32
    yr[q] = flat >> 6;  yc[q] = flat & 63;   // Y tile: 32 x 64
  }

  float4 rx[4], ry[4];
  v8f acc[4] = {};

  // prologue: chunk 0 -> buffer 0
#pragma unroll
  for (int q = 0; q < 4; ++q) {
    rx[q] = *(const float4*)&Xb[(size_t)(bm + xr[q]) * 512 + xc[q]];
    ry[q] = *(const float4*)&Yb[(size_t)yr[q] * 512 + bn + yc[q]];
  }
#pragma unroll
  for (int q = 0; q < 4; ++q) {
    *(float4*)&Xs[0][xr[q] * XSTR + xc[q]] = rx[q];
    *(float4*)&Ys[0][yr[q] * YSTR + yc[q]] = ry[q];
  }
  __syncthreads();

  for (int c = 0; c < 16; ++c) {
    const int cur = c & 1;
    if (c < 15) {
      const int k0 = (c + 1) * 32;
#pragma unroll
      for (int q = 0; q < 4; ++q) {
        rx[q] = *(const float4*)&Xb[(size_t)(bm + xr[q]) * 512 + k0 + xc[q]];
        ry[q] = *(const float4*)&Yb[(size_t)(k0 + yr[q]) * 512 + bn + yc[q]];
      }
    }
    const float* xsc = Xs[cur];
    const float* ysc = Ys[cur];
    const int rowb = (wave * 16 + lp) * XSTR;
#pragma unroll
    for (int k = 0; k < 32; k += 4) {
      const int ka = k + half * 2;
      v2f a;
      a.x = xsc[rowb + ka];
      a.y = xsc[rowb + ka + 1];
#pragma unroll
      for (int j = 0; j < 4; ++j) {
        v2f bf;
        bf.x = ysc[ka * YSTR + j * 16 + lp];
        bf.y = ysc[(ka + 1) * YSTR + j * 16 + lp];
        acc[j] = __builtin_amdgcn_wmma_f32_16x16x4_f32(
            false, a, false, bf, (short)0, acc[j], false, false);
      }
    }
    if (c < 15) {
      const int nxt = (c + 1) & 1;
#pragma unroll
      for (int q = 0; q < 4; ++q) {
        *(float4*)&Xs[nxt][xr[q] * XSTR + xc[q]] = rx[q];
        *(float4*)&Ys[nxt][yr[q] * YSTR + yc[q]] = ry[q];
      }
    }
    __syncthreads();
  }

#pragma unroll
  for (int j = 0; j < 4; ++j)
#pragma unroll
    for (int v = 0; v < 8; ++v) {
      int gm = bm + wave * 16 + v + half * 8;
      int gn = bn + j * 16 + lp;
      float r = acc[j][v];
      if (addX) r += Xb[(size_t)gm * 512 + gn];
      Db[(size_t)gm * 512 + gn] = r;
    }
}

// ---------------- kernel 5: G[b] = C[b] @ B[b]^T  (512x512, K=256) ----------------
#define TSTR 20  // padded stride for 64x16 tiles: lp*20 mod 64 bijective
__global__ __launch_bounds__(128)
void gssd_gemm_nt(const float* __restrict__ Cm, const float* __restrict__ Bm,
                  float* __restrict__ G) {
  __shared__ float Xs[64 * TSTR];
  __shared__ float Ys[64 * TSTR];
  const int t    = threadIdx.x;
  const int wave = t >> 5;
  const int lane = t & 31;
  const int lp   = lane & 15;
  const int half = lane >> 4;
  const int bm = blockIdx.x * 64;
  const int bn = blockIdx.y * 64;
  const int b  = blockIdx.z;
  const float* Xb = Cm + (size_t)b * 512 * 256;
  const float* Yb = Bm + (size_t)b * 512 * 256;
  float* Gb = G + (((size_t)b) << 18);

  v8f acc[4] = {};
  for (int k0 = 0; k0 < 256; k0 += 16) {
#pragma unroll
    for (int q = 0; q < 2; ++q) {
      int flat = (t * 2 + q) * 4;
      int r = flat >> 4, c = flat & 15;
      *(float4*)&Xs[r * TSTR + c] = *(const float4*)&Xb[(size_t)(bm + r) * 256 + k0 + c];
      *(float4*)&Ys[r * TSTR + c] = *(const float4*)&Yb[(size_t)(bn + r) * 256 + k0 + c];
    }
    __syncthreads();
    const int rowb = (wave * 16 + lp) * TSTR;
#pragma unroll
    for (int k = 0; k < 16; k += 4) {
      const int ka = k + half * 2;
      v2f a;
      a.x = Xs[rowb + ka];
      a.y = Xs[rowb + ka + 1];
#pragma unroll
      for (int j = 0; j < 4; ++j) {
        v2f bf;
        bf.x = Ys[(j * 16 + lp) * TSTR + ka];
        bf.y = Ys[(j * 16 + lp) * TSTR + ka + 1];
        acc[j] = __builtin_amdgcn_wmma_f32_16x16x4_f32(
            false, a, false, bf, (short)0, acc[j], false, false);
      }
    }
    __syncthreads();
  }
#pragma unroll
  for (int j = 0; j < 4; ++j)
#pragma unroll
    for (int v = 0; v < 8; ++v) {
      int gm = bm + wave * 16 + v + half * 8;
      int gn = bn + j * 16 + lp;
      Gb[(size_t)gm * 512 + gn] = acc[j][v];
    }
}

// ---- kernel 6: y = (G .* Lm .* dtself_col) @ x + D*x, interleaved store ----
__global__ __launch_bounds__(128)
void gssd_final(const float* __restrict__ G, const float* __restrict__ Lm,
                const float* __restrict__ dself, const float* __restrict__ x,
                const float* __restrict__ Dv, float* __restrict__ out) {
  __shared__ float Ms[64 * TSTR];
  __shared__ float Xv[16 * YSTR];
  const int t    = threadIdx.x;
  const int wave = t >> 5;
  const int lane = t & 31;
  const int lp   = lane & 15;
  const int half = lane >> 4;
  const int bm = blockIdx.x * 64;     // row tile (l)
  const int m  = blockIdx.y;          // b*4 + h
  const int b = m >> 2, h = m & 3;
  const float* Gb = G + (((size_t)b) << 18);
  const float* Lb = Lm + (((size_t)m) << 18);
  const float Dh = Dv[h];

  v8f acc[4] = {};
  for (int t0 = 0; t0 < 512; t0 += 16) {
#pragma unroll
    for (int q = 0; q < 2; ++q) {
      int flat = (t * 2 + q) * 4;
      int r = flat >> 4, c = flat & 15;
      float4 g4 = *(const float4*)&Gb[(size_t)(bm + r) * 512 + t0 + c];
      float4 l4 = *(const float4*)&Lb[(size_t)(bm + r) * 512 + t0 + c];
      int n4 = (((b << 9) + t0 + c) << 2) + h;
      Ms[r * TSTR + c + 0] = g4.x * l4.x * dself[n4];
      Ms[r * TSTR + c + 1] = g4.y * l4.y * dself[n4 + 4];
      Ms[r * TSTR + c + 2] = g4.z * l4.z * dself[n4 + 8];
      Ms[r * TSTR + c + 3] = g4.w * l4.w * dself[n4 + 12];
      int r2 = flat >> 6, c2 = flat & 63;
      *(float4*)&Xv[r2 * YSTR + c2] =
          *(const float4*)&x[(size_t)((b << 9) + t0 + r2) * 256 + h * 64 + c2];
    }
    __syncthreads();
    const int rowb = (wave * 16 + lp) * TSTR;
#pragma unroll
    for (int k = 0; k < 16; k += 4) {
      const int ka = k + half * 2;
      v2f a;
      a.x = Ms[rowb + ka];
      a.y = Ms[rowb + ka + 1];
#pragma unroll
      for (int j = 0; j < 4; ++j) {
        v2f bf;
        bf.x = Xv[ka * YSTR + j * 16 + lp];
        bf.y = Xv[(ka + 1) * YSTR + j * 16 + lp];
        acc[j] = __builtin_amdgcn_wmma_f32_16x16x4_f32(
            false, a, false, bf, (short)0, acc[j], false, false);
      }
    }
    __syncthreads();
  }
#pragma unroll
  for (int j = 0; j < 4; ++j)
#pragma unroll
    for (int v = 0; v < 8; ++v) {
      int gm = bm + wave * 16 + v + half * 8;   // local l
      int gc = j * 16 + lp;                     // head-dim index hd
      int node = (b << 9) + gm;
      float xv = x[(size_t)node * 256 + h * 64 + gc];
      // y.transpose(0,2,3,1): out channel = hd*H + h
      out[(size_t)node * 256 + gc * 4 + h] = acc[j][v] + Dh * xv;
    }
}

// ---------------- host launch ----------------
extern "C" void kernel_launch(void* const* d_in, const int* in_sizes, int n_in,
                              void* d_out, int out_size, void* d_ws, size_t ws_size,
                              hipStream_t stream) {
  (void)in_sizes; (void)n_in; (void)out_size; (void)ws_size;
  const float* x       = (const float*)d_in[0];
  const float* Bmat    = (const float*)d_in[1];
  const float* Cmat    = (const float*)d_in[2];
  const float* dt      = (const float*)d_in[3];
  const float* dt_edge = (const float*)d_in[4];
  const float* dt_bias = (const float*)d_in[5];
  const float* Dv      = (const float*)d_in[6];
  const int*   ei      = (const int*)d_in[7];
  // d_in[8] = data_batch (implied by layout), d_in[9] = diameter (=32, hardcoded)
  float* out = (float*)d_out;

  float* ws    = (float*)d_ws;
  float* a0    = ws;                               // N*H
  float* a1    = a0 + (size_t)N_NODES * NUM_HEADS; // N*H
  float* dum   = a1 + (size_t)N_NODES * NUM_HEADS; // N*H
  float* dself = dum + (size_t)N_NODES * NUM_HEADS;// N*H
  float* W0 = dself + (size_t)N_NODES * NUM_HEADS; // 32 MB each
  float* W1 = W0 + BIGF;
  float* W2 = W1 + BIGF;

  // adj (in W2) must start zeroed for the edge scatter
  hipMemsetAsync(W2, 0, BIGF * sizeof(float), stream);

  gssd_prep<<<(N_NODES * NUM_HEADS + 255) / 256, 256, 0, stream>>>(dt, dt_bias, a0, a1, dum, dself);
  gssd_edges<<<(N_EDGES + 255) / 256, 256, 0, stream>>>(dt_edge, dt_bias, ei, a0, a1, W2);
  gssd_normT<<<dim3(512, NMAT), 128, 0, stream>>>(W2, dum, W0 /*AT = P0*/, W1 /*Q0 = I+AT*/);

  // Neumann doubling chain (diameter 32 -> 4 doubling steps):
  //   P <- P@P ; Q <- Q@P + Q      (3-buffer rotation; P_old dies before Q_new write)
  dim3 gg(8, 8, NMAT);
  gssd_gemm512<<<gg, 128, 0, stream>>>(W0, W0, W2, 0);  // P1 = P0^2      -> W2
  gssd_gemm512<<<gg, 128, 0, stream>>>(W1, W2, W0, 1);  // Q1 = Q0@P1+Q0  -> W0
  gssd_gemm512<<<gg, 128, 0, stream>>>(W2, W2, W1, 0);  // P2             -> W1
  gssd_gemm512<<<gg, 128, 0, stream>>>(W0, W1, W2, 1);  // Q2             -> W2
  gssd_gemm512<<<gg, 128, 0, stream>>>(W1, W1, W0, 0);  // P3             -> W0
  gssd_gemm512<<<gg, 128, 0, stream>>>(W2, W0, W1, 1);  // Q3             -> W1
  gssd_gemm512<<<gg, 128, 0, stream>>>(W0, W0, W2, 0);  // P4             -> W2
  gssd_gemm512<<<gg, 128, 0, stream>>>(W1, W2, W0, 1);  // Q4 = Lm        -> W0

  // G[b] = C_b @ B_b^T (head-independent), into free buffer W1
  gssd_gemm_nt<<<dim3(8, 8, NUM_GRAPHS), 128, 0, stream>>>(Cmat, Bmat, W1);

  // y = (G .* Lm .* dt_self) @ x + D*x, with (hd,h) channel interleave
  gssd_final<<<dim3(8, NMAT), 128, 0, stream>>>(W1, W0, dself, x, Dv, out);
}